// LSTM_31688268710511
// MI455X (gfx1250) — compile-verified
//
#include <hip/hip_runtime.h>

// Fused 2-layer LSTM + FC head for MI455X (gfx1250, wave32, WMMA).
// One workgroup = 16 batch rows, persistent across all T=512 steps.
// Weights held in registers as V_WMMA_F32_16X16X4_F32 B-fragments.

typedef float v2f __attribute__((ext_vector_type(2)));
typedef float v8f __attribute__((ext_vector_type(8)));

#define B_TOT   2048
#define T_LEN   512
#define I_DIM   9
#define H_DIM   56
#define G_DIM   224     // 4*H
#define HSTR    58      // LDS row stride for h tiles (bank-conflict padding)
#define XSTR    12      // x tile K padded 9 -> 12 (3 WMMA k-tiles)
#define GSTR    228     // LDS row stride for gates tile
#define BTILE   16
#define NTHREADS 448    // 14 waves; wave w owns gate columns [16w, 16w+16)

__device__ __forceinline__ float fast_sigmoid(float v) {
  return 1.0f / (1.0f + __expf(-v));
}
__device__ __forceinline__ float fast_tanh(float v) {
  return 1.0f - 2.0f / (1.0f + __expf(2.0f * v));
}

__global__ __launch_bounds__(NTHREADS, 1) void lstm_fused_kernel(
    const float* __restrict__ x,
    const float* __restrict__ w_ih0, const float* __restrict__ w_hh0,
    const float* __restrict__ b_ih0, const float* __restrict__ b_hh0,
    const float* __restrict__ w_ih1, const float* __restrict__ w_hh1,
    const float* __restrict__ b_ih1, const float* __restrict__ b_hh1,
    const float* __restrict__ fc1_w, const float* __restrict__ fc1_b,
    const float* __restrict__ fc2_w, const float* __restrict__ fc2_b,
    float* __restrict__ out)
{
  __shared__ float h0_lds[BTILE * HSTR];
  __shared__ float h1_lds[BTILE * HSTR];
  __shared__ float x_lds [BTILE * XSTR];
  __shared__ float g_lds [BTILE * GSTR];

  const int tid   = threadIdx.x;
  const int lane  = tid & 31;
  const int wave  = tid >> 5;            // 0..13
  const int nloc  = lane & 15;           // N within 16-tile / M for A-frag
  const int koff  = (lane >> 4) << 1;    // 0 or 2 (K pair select)
  const int mhalf = (lane >> 4) << 3;    // 0 or 8 (C/D row half)
  const int bbase = blockIdx.x * BTILE;  // batch tile base

  // zero h/c state and x padding
  for (int i = tid; i < BTILE * HSTR; i += NTHREADS) { h0_lds[i] = 0.f; h1_lds[i] = 0.f; }
  for (int i = tid; i < BTILE * XSTR; i += NTHREADS) x_lds[i] = 0.f;

  float c0[2] = {0.f, 0.f};
  float c1[2] = {0.f, 0.f};

  // elementwise cell mapping: 448 threads x 2 rows = 16*56 cells
  const int col = tid % H_DIM;           // hidden column 0..55
  const int m2  = (tid / H_DIM) * 2;     // row pair base 0,2,...,14

  // fused biases (b_ih + b_hh), gate order i,f,g,o
  const float bi0 = b_ih0[0*H_DIM+col] + b_hh0[0*H_DIM+col];
  const float bf0 = b_ih0[1*H_DIM+col] + b_hh0[1*H_DIM+col];
  const float bg0 = b_ih0[2*H_DIM+col] + b_hh0[2*H_DIM+col];
  const float bo0 = b_ih0[3*H_DIM+col] + b_hh0[3*H_DIM+col];
  const float bi1 = b_ih1[0*H_DIM+col] + b_hh1[0*H_DIM+col];
  const float bf1 = b_ih1[1*H_DIM+col] + b_hh1[1*H_DIM+col];
  const float bg1 = b_ih1[2*H_DIM+col] + b_hh1[2*H_DIM+col];
  const float bo1 = b_ih1[3*H_DIM+col] + b_hh1[3*H_DIM+col];

  // ---- load persistent weight B-fragments into registers ----
  // B layout (4x16 f32): lane l: n = l%16, v.x = w^T[k+koff][n], v.y = w^T[k+koff+1][n]
  v2f whh0f[14], whh1f[14], wih1f[14], wih0f[3];
  {
    const int gc = wave * 16 + nloc;     // gate column 0..223
#pragma unroll
    for (int kt = 0; kt < 14; ++kt) {
      const int k = kt * 4 + koff;       // even -> 8B aligned (H_DIM even)
      whh0f[kt] = *(const v2f*)(w_hh0 + gc * H_DIM + k);
      whh1f[kt] = *(const v2f*)(w_hh1 + gc * H_DIM + k);
      wih1f[kt] = *(const v2f*)(w_ih1 + gc * H_DIM + k);
    }
#pragma unroll
    for (int kt = 0; kt < 3; ++kt) {     // K=9 padded to 12 with zeros
      const int k = kt * 4 + koff;
      v2f v;
      v.x = (k     < I_DIM) ? w_ih0[gc * I_DIM + k]     : 0.f;
      v.y = (k + 1 < I_DIM) ? w_ih0[gc * I_DIM + k + 1] : 0.f;
      wih0f[kt] = v;
    }
  }

  __syncthreads();

  for (int t = 0; t < T_LEN; ++t) {
    // ---- A: stage x_t tile [16 x 9] into LDS (padding cols stay 0) ----
    if (tid < BTILE * I_DIM) {
      const int m = tid / I_DIM, ii = tid % I_DIM;
      const size_t base = (size_t)(bbase + m) * (T_LEN * I_DIM) + (size_t)t * I_DIM + ii;
      x_lds[m * XSTR + ii] = x[base];
      if (t + 1 < T_LEN) __builtin_prefetch(x + base + I_DIM, 0, 0);
    }
    __syncthreads();

    // ---- B: layer0 gates = x_t @ w_ih0^T + h0 @ w_hh0^T ----
    {
      v8f acc = {};
#pragma unroll
      for (int kt = 0; kt < 3; ++kt) {
        v2f a = *(const v2f*)(x_lds + nloc * XSTR + kt * 4 + koff);
        acc = __builtin_amdgcn_wmma_f32_16x16x4_f32(false, a, false, wih0f[kt],
                                                    (short)0, acc, false, false);
      }
#pragma unroll
      for (int kt = 0; kt < 14; ++kt) {
        v2f a = *(const v2f*)(h0_lds + nloc * HSTR + kt * 4 + koff);
        acc = __builtin_amdgcn_wmma_f32_16x16x4_f32(false, a, false, whh0f[kt],
                                                    (short)0, acc, false, false);
      }
#pragma unroll
      for (int r = 0; r < 8; ++r)
        g_lds[(r + mhalf) * GSTR + wave * 16 + nloc] = acc[r];
    }
    __syncthreads();

    // ---- C: layer0 cell update ----
#pragma unroll
    for (int q = 0; q < 2; ++q) {
      const int m = m2 + q;
      const float gi = fast_sigmoid(g_lds[m * GSTR + 0 * H_DIM + col] + bi0);
      const float gf = fast_sigmoid(g_lds[m * GSTR + 1 * H_DIM + col] + bf0);
      const float gg = fast_tanh   (g_lds[m * GSTR + 2 * H_DIM + col] + bg0);
      const float go = fast_sigmoid(g_lds[m * GSTR + 3 * H_DIM + col] + bo0);
      c0[q] = gf * c0[q] + gi * gg;
      h0_lds[m * HSTR + col] = go * fast_tanh(c0[q]);
    }
    __syncthreads();

    // ---- D: layer1 gates = h0_t @ w_ih1^T + h1 @ w_hh1^T ----
    {
      v8f acc = {};
#pragma unroll
      for (int kt = 0; kt < 14; ++kt) {
        v2f a = *(const v2f*)(h0_lds + nloc * HSTR + kt * 4 + koff);
        acc = __builtin_amdgcn_wmma_f32_16x16x4_f32(false, a, false, wih1f[kt],
                                                    (short)0, acc, false, false);
      }
#pragma unroll
      for (int kt = 0; kt < 14; ++kt) {
        v2f a = *(const v2f*)(h1_lds + nloc * HSTR + kt * 4 + koff);
        acc = __builtin_amdgcn_wmma_f32_16x16x4_f32(false, a, false, whh1f[kt],
                                                    (short)0, acc, false, false);
      }
#pragma unroll
      for (int r = 0; r < 8; ++r)
        g_lds[(r + mhalf) * GSTR + wave * 16 + nloc] = acc[r];
    }
    __syncthreads();

    // ---- E: layer1 cell update (barrier at loop end covers the hazards) ----
#pragma unroll
    for (int q = 0; q < 2; ++q) {
      const int m = m2 + q;
      const float gi = fast_sigmoid(g_lds[m * GSTR + 0 * H_DIM + col] + bi1);
      const float gf = fast_sigmoid(g_lds[m * GSTR + 1 * H_DIM + col] + bf1);
      const float gg = fast_tanh   (g_lds[m * GSTR + 2 * H_DIM + col] + bg1);
      const float go = fast_sigmoid(g_lds[m * GSTR + 3 * H_DIM + col] + bo1);
      c1[q] = gf * c1[q] + gi * gg;
      h1_lds[m * HSTR + col] = go * fast_tanh(c1[q]);
    }
    __syncthreads();
  }

  // ---- FC head on final h1: y = relu(h1 @ fc1^T + b1) @ fc2^T + b2 ----
  if (tid < BTILE) {
    float acc2 = fc2_b[0];
    for (int j = 0; j < H_DIM / 2; ++j) {
      float a = fc1_b[j];
#pragma unroll 8
      for (int k = 0; k < H_DIM; ++k)
        a = fmaf(h1_lds[tid * HSTR + k], fc1_w[j * H_DIM + k], a);
      a = fmaxf(a, 0.f);
      acc2 = fmaf(a, fc2_w[j], acc2);
    }
    out[bbase + tid] = acc2;
  }
}

extern "C" void kernel_launch(void* const* d_in, const int* in_sizes, int n_in,
                              void* d_out, int out_size, void* d_ws, size_t ws_size,
                              hipStream_t stream) {
  const float* x     = (const float*)d_in[0];
  const float* w_ih0 = (const float*)d_in[1];
  const float* w_hh0 = (const float*)d_in[2];
  const float* b_ih0 = (const float*)d_in[3];
  const float* b_hh0 = (const float*)d_in[4];
  const float* w_ih1 = (const float*)d_in[5];
  const float* w_hh1 = (const float*)d_in[6];
  const float* b_ih1 = (const float*)d_in[7];
  const float* b_hh1 = (const float*)d_in[8];
  const float* fc1_w = (const float*)d_in[9];
  const float* fc1_b = (const float*)d_in[10];
  const float* fc2_w = (const float*)d_in[11];
  const float* fc2_b = (const float*)d_in[12];
  float* out = (float*)d_out;

  dim3 grid(B_TOT / BTILE);   // 128 workgroups
  dim3 block(NTHREADS);       // 448 threads = 14 waves
  hipLaunchKernelGGL(lstm_fused_kernel, grid, block, 0, stream,
                     x, w_ih0, w_hh0, b_ih0, b_hh0,
                     w_ih1, w_hh1, b_ih1, b_hh1,
                     fc1_w, fc1_b, fc2_w, fc2_b, out);
}